// CustomCNN_61649960566889
// MI455X (gfx1250) — compile-verified
//
#include <hip/hip_runtime.h>
#include <hip/hip_bf16.h>
#include <math.h>

typedef __attribute__((ext_vector_type(2))) float v2f;
typedef __attribute__((ext_vector_type(8))) float v8f;

#define WAVES_PER_BLOCK 4
#define IMGS_PER_WAVE   16
#define KPAD            52   // 49 padded up to a multiple of 4 (zero fill)

// Wave-local LDS sync: each wave uses a private LDS region, so a split-counter
// DS wait (CDNA5 s_wait_dscnt) is sufficient for cross-lane visibility.
__device__ __forceinline__ void wsync() {
    __builtin_amdgcn_wave_barrier();
    asm volatile("s_wait_dscnt 0" ::: "memory");
    __builtin_amdgcn_wave_barrier();
}

__global__ __launch_bounds__(128, 1)
void cnn_fused_kernel(const float* __restrict__ x,
                      const float* __restrict__ k1,
                      const float* __restrict__ k2,
                      const float* __restrict__ fc1w,
                      const float* __restrict__ fc1b,
                      const float* __restrict__ fc2w,
                      const float* __restrict__ fc2b,
                      float* __restrict__ out,
                      int B) {
    __shared__ float s_img[WAVES_PER_BLOCK][784];                 // 28x28 stage
    __shared__ float s_p1 [WAVES_PER_BLOCK][196];                 // 14x14 pooled
    __shared__ float s_act[WAVES_PER_BLOCK][IMGS_PER_WAVE * KPAD];// [16 x 52]
    __shared__ float s_h  [WAVES_PER_BLOCK][IMGS_PER_WAVE * 128]; // [16 x 128]

    const int lane = threadIdx.x & 31;
    const int wave = threadIdx.x >> 5;
    const long tile = (long)blockIdx.x * WAVES_PER_BLOCK + wave;  // 16-image tile
    const long base = tile * IMGS_PER_WAVE;
    if (base >= B) return;   // wave-uniform; no block barriers used below

    float* img = s_img[wave];
    float* p1  = s_p1[wave];
    float* act = s_act[wave];
    float* h   = s_h[wave];

    // 5x5 kernels into registers (wave-uniform -> scalar loads)
    float ck1[25], ck2[25];
#pragma unroll
    for (int i = 0; i < 25; ++i) { ck1[i] = k1[i]; ck2[i] = k2[i]; }

    // ---------------- conv stages, one image at a time ----------------
    for (int im = 0; im < IMGS_PER_WAVE; ++im) {
        long gi = base + im; if (gi > B - 1) gi = B - 1;          // clamp partial tile
        const float* src = x + gi * 784;
        for (int i = lane; i < 784; i += 32) img[i] = src[i];
        if (im + 1 < IMGS_PER_WAVE) {
            long gn = base + im + 1; if (gn > B - 1) gn = B - 1;
            __builtin_prefetch(x + gn * 784 + lane * 24, 0, 0);   // global_prefetch
        }
        wsync();

        // conv1(5x5, pad2) + relu + 2x2 maxpool  -> p1[14x14]
        for (int o = lane; o < 196; o += 32) {
            int py = o / 14, px = o % 14;
            float m = 0.0f;                    // relu>=0, so 0 is a valid identity
#pragma unroll
            for (int dy = 0; dy < 2; ++dy)
#pragma unroll
            for (int dx = 0; dx < 2; ++dx) {
                int y = py * 2 + dy, xx = px * 2 + dx;
                float acc = 0.f;
#pragma unroll
                for (int ky = 0; ky < 5; ++ky) {
                    int iy = y + ky - 2;
                    if (iy < 0 || iy >= 28) continue;
#pragma unroll
                    for (int kx = 0; kx < 5; ++kx) {
                        int ix = xx + kx - 2;
                        if (ix < 0 || ix >= 28) continue;
                        acc = fmaf(img[iy * 28 + ix], ck1[ky * 5 + kx], acc);
                    }
                }
                m = fmaxf(m, acc);
            }
            p1[o] = m;
        }
        wsync();

        // conv2(5x5, pad2) + relu + 2x2 maxpool -> act[im][0..48]
        for (int o = lane; o < 49; o += 32) {
            int py = o / 7, px = o % 7;
            float m = 0.0f;
#pragma unroll
            for (int dy = 0; dy < 2; ++dy)
#pragma unroll
            for (int dx = 0; dx < 2; ++dx) {
                int y = py * 2 + dy, xx = px * 2 + dx;
                float acc = 0.f;
#pragma unroll
                for (int ky = 0; ky < 5; ++ky) {
                    int iy = y + ky - 2;
                    if (iy < 0 || iy >= 14) continue;
#pragma unroll
                    for (int kx = 0; kx < 5; ++kx) {
                        int ix = xx + kx - 2;
                        if (ix < 0 || ix >= 14) continue;
                        acc = fmaf(p1[iy * 14 + ix], ck2[ky * 5 + kx], acc);
                    }
                }
                m = fmaxf(m, acc);
            }
            act[im * KPAD + o] = m;
        }
        if (lane < KPAD - 49) act[im * KPAD + 49 + lane] = 0.f;   // zero K padding
        wsync();
    }

    // ---------------- FC1: [16x49] @ [49x128] via v_wmma_f32_16x16x4_f32 ----
    // f32 A layout (16x4): lanes 0-15 hold K={0,1}, lanes 16-31 hold K={2,3}.
    // B layout (4x16): VGPR0 holds rows K=0 (lanes 0-15) / K=2 (lanes 16-31),
    //                  VGPR1 holds rows K=1 / K=3; column N = lane&15.
    const int row   = lane & 15;            // A-row (image) and B/D column index
    const int khalf = (lane >> 4) << 1;     // 0 or 2
    const int mbase = (lane >> 4) << 3;     // C/D: M = r (+8 for hi lanes)

    for (int nt = 0; nt < 8; ++nt) {
        v8f acc = {};
#pragma unroll
        for (int k = 0; k < KPAD / 4; ++k) {
            int kb = k * 4 + khalf;
            v2f a, b;
            a.x = act[row * KPAD + kb];
            a.y = act[row * KPAD + kb + 1];
            int n = nt * 16 + row;
            b.x = (kb     < 49) ? fc1w[n * 49 + kb]     : 0.f;
            b.y = (kb + 1 < 49) ? fc1w[n * 49 + kb + 1] : 0.f;
            acc = __builtin_amdgcn_wmma_f32_16x16x4_f32(
                      false, a, false, b, (short)0, acc, false, false);
        }
        int n = nt * 16 + row;
        float bias = fc1b[n];
#pragma unroll
        for (int r = 0; r < 8; ++r) {
            int m = r + mbase;
            h[m * 128 + n] = fmaxf(acc[r] + bias, 0.f);
        }
    }
    wsync();

    // ---------------- FC2: [16x128] @ [128x10(pad16)] ----------------
    v8f acc2 = {};
#pragma unroll
    for (int k = 0; k < 32; ++k) {
        int kb = k * 4 + khalf;
        v2f a, b;
        a.x = h[row * 128 + kb];
        a.y = h[row * 128 + kb + 1];
        b.x = (row < 10) ? fc2w[row * 128 + kb]     : 0.f;
        b.y = (row < 10) ? fc2w[row * 128 + kb + 1] : 0.f;
        acc2 = __builtin_amdgcn_wmma_f32_16x16x4_f32(
                   false, a, false, b, (short)0, acc2, false, false);
    }
    // logits -> reuse act region as [16 rows x 16 cols]
    float* lg = act;
    {
        float bias = (row < 10) ? fc2b[row] : 0.f;
#pragma unroll
        for (int r = 0; r < 8; ++r) {
            int m = r + mbase;
            lg[m * 16 + row] = acc2[r] + bias;
        }
    }
    wsync();

    // ---------------- log_softmax over 10 classes, write out ----------------
    for (int idx = lane; idx < 160; idx += 32) {
        int m = idx / 10, n = idx % 10;
        if (base + m >= B) continue;
        float mx = -1e30f;
#pragma unroll
        for (int j = 0; j < 10; ++j) mx = fmaxf(mx, lg[m * 16 + j]);
        float s = 0.f;
#pragma unroll
        for (int j = 0; j < 10; ++j) s += expf(lg[m * 16 + j] - mx);
        out[(base + m) * 10 + n] = lg[m * 16 + n] - mx - logf(s);
    }
}

extern "C" void kernel_launch(void* const* d_in, const int* in_sizes, int n_in,
                              void* d_out, int out_size, void* d_ws, size_t ws_size,
                              hipStream_t stream) {
    const float* x    = (const float*)d_in[0];
    const float* k1   = (const float*)d_in[1];
    const float* k2   = (const float*)d_in[2];
    const float* fc1w = (const float*)d_in[3];
    const float* fc1b = (const float*)d_in[4];
    const float* fc2w = (const float*)d_in[5];
    const float* fc2b = (const float*)d_in[6];
    float* out = (float*)d_out;

    int B = in_sizes[0] / 784;                          // 65536
    int tiles  = (B + IMGS_PER_WAVE - 1) / IMGS_PER_WAVE;
    int blocks = (tiles + WAVES_PER_BLOCK - 1) / WAVES_PER_BLOCK;

    cnn_fused_kernel<<<blocks, WAVES_PER_BLOCK * 32, 0, stream>>>(
        x, k1, k2, fc1w, fc1b, fc2w, fc2b, out, B);
}